// SDF_Loss_Interpolated_27273042330167
// MI455X (gfx1250) — compile-verified
//
#include <hip/hip_runtime.h>
#include <hip/hip_bf16.h>
#include <stdint.h>

#ifndef __has_builtin
#define __has_builtin(x) 0
#endif

#define GRID_W      4096
#define GRID_MAX    4095

constexpr int BLOCK = 256;   // 8 waves (wave32)
constexpr int PPT   = 4;     // points per thread per tile
constexpr int TILE  = BLOCK * PPT;  // 1024 points / tile
constexpr int MAX_BLOCKS = 2048;

typedef int v4i __attribute__((ext_vector_type(4)));
typedef __attribute__((address_space(1))) v4i as1_v4i;   // global (__device__)
typedef __attribute__((address_space(3))) v4i as3_v4i;   // LDS (__shared__)

// ---- CDNA5 async global->LDS copy (ASYNCcnt path), 16B per lane ----
__device__ __forceinline__ void async_copy_b128(const void* gsrc, void* ldst) {
#if __has_builtin(__builtin_amdgcn_global_load_async_to_lds_b128)
    // (global int4*, lds int4*, imm offset, imm cpol)
    __builtin_amdgcn_global_load_async_to_lds_b128(
        (as1_v4i*)(uintptr_t)gsrc,                 // int->ptr: numeric identity for AS1
        (as3_v4i*)(uint32_t)(uintptr_t)ldst,       // low 32 bits of flat == LDS offset
        0, 0);
#else
    uint32_t lds_off = (uint32_t)(uintptr_t)ldst;
    asm volatile("global_load_async_to_lds_b128 %0, %1, off"
                 :: "v"(lds_off), "v"(gsrc) : "memory");
#endif
}

__device__ __forceinline__ void wait_async_all() {
#if __has_builtin(__builtin_amdgcn_s_wait_asynccnt)
    __builtin_amdgcn_s_wait_asynccnt(0);
#else
    asm volatile("s_wait_asynccnt 0" ::: "memory");
#endif
}

__global__ __launch_bounds__(BLOCK)
void sdf_bilerp_kernel(const float* __restrict__ x,
                       const float* __restrict__ y,
                       const float* __restrict__ sdf,
                       float* __restrict__ out_vals,   // d_out + 1
                       float* __restrict__ partial,    // d_ws, one per block
                       int n) {
    __shared__ float stage[2][2][TILE];   // [buf][x/y][point] = 16 KB
    __shared__ float wsum[BLOCK / 32];

    const int tid    = threadIdx.x;
    const int lane   = tid & 31;
    const int wid    = tid >> 5;
    const int nTiles = (n + TILE - 1) / TILE;
    const int stride = gridDim.x;

    // Each wave stages its own 128-point slice of x and y (2 async b128/wave).
    auto stage_tile = [&](int tb, int buf) {
        long base = (long)tb * TILE;
        long mx   = (long)n - TILE; if (mx < 0) mx = 0;
        if (base > mx) base = mx;                  // clamp: never read past end
        const int off = wid * 128 + lane * 4;      // 4 floats = 16B per lane
        async_copy_b128(x + base + off, &stage[buf][0][off]);
        async_copy_b128(y + base + off, &stage[buf][1][off]);
    };

    const int tb0 = (int)blockIdx.x;
    if (tb0 < nTiles) stage_tile(tb0, 0);
    wait_async_all();
    __syncthreads();

    float sum = 0.f;
    int cur = 0;
    for (int tb = tb0; tb < nTiles; tb += stride) {
        const int nb = tb + stride;
        if (nb < nTiles) stage_tile(nb, cur ^ 1);  // prefetch overlaps compute

        const long base = (long)tb * TILE;
        #pragma unroll
        for (int k = 0; k < PPT; ++k) {
            const int  idx32 = tid + k * BLOCK;    // coalesced per-lane b32 stores
            const long idx   = base + idx32;
            if (idx < n) {
                const float xf = stage[cur][0][idx32];
                const float yf = stage[cur][1][idx32];
                const int x1 = (int)floorf(xf);
                const int y1 = (int)floorf(yf);
                const int x1c = min(max(x1, 0), GRID_MAX);
                const int x2c = min(max(x1 + 1, 0), GRID_MAX);
                const int y1c = min(max(y1, 0), GRID_MAX);
                const int y2c = min(max(y1 + 1, 0), GRID_MAX);
                const float t = xf - (float)x1c;
                const float u = yf - (float)y1c;
                // 32-bit index math: 4095*4096+4095 < 2^32
                const uint32_t r1 = (uint32_t)x1c << 12;
                const uint32_t r2 = (uint32_t)x2c << 12;
                const float v11 = sdf[r1 + (uint32_t)y1c];
                const float v21 = sdf[r2 + (uint32_t)y1c];
                const float v12 = sdf[r1 + (uint32_t)y2c];
                const float v22 = sdf[r2 + (uint32_t)y2c];
                float a = fmaf(t, v21 - v11, v11);
                float b = fmaf(t, v22 - v12, v12);
                float v = fmaf(u, b - a, a);
                v = fminf(fmaxf(v, -1000.f), 0.f);
                __builtin_nontemporal_store(v, out_vals + idx);  // don't pollute L2
                sum += v;
            }
        }
        wait_async_all();   // own next-tile async ops complete
        __syncthreads();    // all waves' slices complete; cur buffer free to refill
        cur ^= 1;
    }

    // wave32 reduction, then cross-wave via LDS
    #pragma unroll
    for (int off = 16; off > 0; off >>= 1)
        sum += __shfl_xor(sum, off, 32);
    if (lane == 0) wsum[wid] = sum;
    __syncthreads();
    if (tid == 0) {
        float s = 0.f;
        #pragma unroll
        for (int w = 0; w < BLOCK / 32; ++w) s += wsum[w];
        partial[blockIdx.x] = s;
    }
}

__global__ __launch_bounds__(BLOCK)
void sdf_reduce_kernel(const float* __restrict__ partial, int nb,
                       float* __restrict__ loss) {
    __shared__ float wsum[BLOCK / 32];
    float s = 0.f;
    for (int i = threadIdx.x; i < nb; i += BLOCK) s += partial[i];
    #pragma unroll
    for (int off = 16; off > 0; off >>= 1)
        s += __shfl_xor(s, off, 32);
    const int lane = threadIdx.x & 31, wid = threadIdx.x >> 5;
    if (lane == 0) wsum[wid] = s;
    __syncthreads();
    if (threadIdx.x == 0) {
        float t = 0.f;
        #pragma unroll
        for (int w = 0; w < BLOCK / 32; ++w) t += wsum[w];
        *loss = t;   // deterministic scalar loss
    }
}

extern "C" void kernel_launch(void* const* d_in, const int* in_sizes, int n_in,
                              void* d_out, int out_size, void* d_ws, size_t ws_size,
                              hipStream_t stream) {
    const float* x   = (const float*)d_in[0];
    const float* y   = (const float*)d_in[1];
    const float* sdf = (const float*)d_in[2];
    float* out = (float*)d_out;          // out[0] = loss, out[1..] = sdf_values
    const int n = in_sizes[0];

    const int nTiles = (n + TILE - 1) / TILE;
    int blocks = nTiles < MAX_BLOCKS ? nTiles : MAX_BLOCKS;
    if ((size_t)blocks * sizeof(float) > ws_size) {
        size_t cap = ws_size / sizeof(float);
        blocks = cap < 1 ? 1 : (int)cap;
    }
    float* partial = (float*)d_ws;

    sdf_bilerp_kernel<<<blocks, BLOCK, 0, stream>>>(x, y, sdf, out + 1, partial, n);
    sdf_reduce_kernel<<<1, BLOCK, 0, stream>>>(partial, blocks, out);
}